// LightGCN_18502719111519
// MI455X (gfx1250) — compile-verified
//
#include <hip/hip_runtime.h>
#include <cstdint>
#include <cstddef>

#define DIM     64
#define TILE    512     // edges staged per block
#define BLOCK   256     // 8 waves (wave32)

typedef float float4v __attribute__((ext_vector_type(4)));

// ---- gfx1250 async global->LDS staging (guarded; fallback = plain LDS fill) ----
#if __has_builtin(__builtin_amdgcn_global_load_async_to_lds_b64)
#define HAVE_ASYNC 1
// builtin signature (from hipcc diagnostic): (v2i as1* src, v2i as3* dst, Ii off, Ii cpol)
typedef int v2i __attribute__((__vector_size__(8)));
typedef __attribute__((address_space(1))) v2i g_v2i;
typedef __attribute__((address_space(3))) v2i l_v2i;
#define TO_GBL64(p) ((g_v2i*)(uintptr_t)(p))
#define TO_LDS64(p) ((l_v2i*)(uint32_t)(uintptr_t)(p))
#else
#define HAVE_ASYNC 0
#endif

__device__ __forceinline__ void wait_async_zero() {
#if __has_builtin(__builtin_amdgcn_s_wait_asynccnt)
    __builtin_amdgcn_s_wait_asynccnt(0);
#elif HAVE_ASYNC
    asm volatile("s_wait_asynccnt 0" ::: "memory");
#endif
}

// ---------------------------------------------------------------------------
// SpMM stage: nxt[row[e]] += vals[e] * cur[col[e]]  (edge-parallel).
// A block stages TILE edges' metadata into LDS (async on gfx1250), then
// 16 lanes cooperate per edge: lane L handles dims [4L, 4L+3] as one float4
// gather (global_load_b128) and 4 hardware f32 atomic adds.
// (Placed first in the file so the disasm snippet shows this kernel.)
// ---------------------------------------------------------------------------
__global__ void lgcn_spmm(const float* __restrict__ cur, float* __restrict__ nxt,
                          const float* __restrict__ vals, const int* __restrict__ rows,
                          const int* __restrict__ cols, int E) {
    __shared__ __align__(16) int   s_row[TILE];
    __shared__ __align__(16) int   s_col[TILE];
    __shared__ __align__(16) float s_val[TILE];

    const int       t    = threadIdx.x;
    const long long base = (long long)blockIdx.x * TILE;

#if HAVE_ASYNC
    if (base + TILE <= (long long)E) {
        // 256 threads x 8B x 3 arrays covers the whole 6 KB tile
        const int i = t * 2;
        __builtin_amdgcn_global_load_async_to_lds_b64(TO_GBL64(rows + base + i),
                                                      TO_LDS64(&s_row[i]), 0, 0);
        __builtin_amdgcn_global_load_async_to_lds_b64(TO_GBL64(cols + base + i),
                                                      TO_LDS64(&s_col[i]), 0, 0);
        __builtin_amdgcn_global_load_async_to_lds_b64(TO_GBL64(vals + base + i),
                                                      TO_LDS64(&s_val[i]), 0, 0);
        wait_async_zero();
        __syncthreads();
    } else
#endif
    {
        for (int i = t; i < TILE; i += BLOCK) {
            long long e = base + i;
            if (e < E) { s_row[i] = rows[e]; s_col[i] = cols[e]; s_val[i] = vals[e]; }
        }
        __syncthreads();
    }

    const int lane16 = t & 15;   // which 4-dim chunk of the 64-dim row
    const int eslot  = t >> 4;   // which edge within a 16-edge stripe
    const float4v* cur4 = (const float4v*)cur;

    long long rem = (long long)E - base;
    const int tileCount = (rem < (long long)TILE) ? (int)rem : TILE;

    for (int i = eslot; i < tileCount; i += 16) {
        const int   r = s_row[i];        // LDS broadcast: 16 lanes, same address
        const int   c = s_col[i];
        const float v = s_val[i];
        float4v x = cur4[(size_t)c * (DIM / 4) + lane16];   // global_load_b128, L2-resident
        float4v m = x * v;
        float* dst = nxt + (size_t)r * DIM + lane16 * 4;
        // non-returning hardware f32 atomics (no CAS loop)
        unsafeAtomicAdd(dst + 0, m.x);
        unsafeAtomicAdd(dst + 1, m.y);
        unsafeAtomicAdd(dst + 2, m.z);
        unsafeAtomicAdd(dst + 3, m.w);
    }
}

// ---------------------------------------------------------------------------
// init: out_emb0 = concat(ue, ie); out_mean = emb0 (acc starts at emb0);
//       bufA = emb0 (cur); bufB = 0 (first scatter target).
// ---------------------------------------------------------------------------
__global__ void lgcn_init(const float* __restrict__ ue, const float* __restrict__ ie,
                          float* __restrict__ out_mean, float* __restrict__ out_emb0,
                          float* __restrict__ bufA, float* __restrict__ bufB,
                          int userElems, int totalElems) {
    int i4  = blockIdx.x * blockDim.x + threadIdx.x;
    int idx = i4 * 4;
    if (idx >= totalElems) return;
    float4v v = (idx < userElems) ? *(const float4v*)(ue + idx)
                                  : *(const float4v*)(ie + (idx - userElems));
    // write-once output: stream past L2 to protect the resident working set
    __builtin_nontemporal_store(v, (float4v*)(out_emb0 + idx));
    *(float4v*)(out_mean + idx) = v;   // re-read every stage -> keep cacheable
    *(float4v*)(bufA + idx)     = v;
    float4v z = {0.f, 0.f, 0.f, 0.f};
    *(float4v*)(bufB + idx)     = z;
}

// ---------------------------------------------------------------------------
// accumulate: acc += nxt; zero the buffer that becomes the next scatter
// target (fused to save a pass). Final stage: acc = (acc + nxt) * 0.25,
// streamed out non-temporally.
// ---------------------------------------------------------------------------
__global__ void lgcn_accum(float* __restrict__ acc, const float* __restrict__ nxt,
                           float* __restrict__ zbuf, int total4, int isFinal) {
    int i = blockIdx.x * blockDim.x + threadIdx.x;
    if (i >= total4) return;
    float4v a = ((const float4v*)acc)[i];
    float4v n = ((const float4v*)nxt)[i];
    a = a + n;
    if (isFinal) {
        a = a * 0.25f;
        __builtin_nontemporal_store(a, ((float4v*)acc) + i);
    } else {
        ((float4v*)acc)[i] = a;
        float4v z = {0.f, 0.f, 0.f, 0.f};
        ((float4v*)zbuf)[i] = z;
    }
}

// ---------------------------------------------------------------------------
extern "C" void kernel_launch(void* const* d_in, const int* in_sizes, int n_in,
                              void* d_out, int out_size, void* d_ws, size_t ws_size,
                              hipStream_t stream) {
    const float* ue   = (const float*)d_in[0];
    const float* ie   = (const float*)d_in[1];
    const float* vals = (const float*)d_in[2];
    const int*   rows = (const int*)d_in[3];
    const int*   cols = (const int*)d_in[4];
    // d_in[5] = stages (compile-time 3 in the reference; unrolled host-side)

    const int userElems  = in_sizes[0];
    const int totalElems = in_sizes[0] + in_sizes[1];   // N * 64
    const int E          = in_sizes[2];

    float* out_mean = (float*)d_out;
    float* out_emb0 = out_mean + (size_t)totalElems;

    float* bufA = (float*)d_ws;                 // cur   (25.6 MB)
    float* bufB = bufA + (size_t)totalElems;    // next  (25.6 MB)

    const int total4   = totalElems / 4;
    const int gridNode = (total4 + BLOCK - 1) / BLOCK;
    const int gridEdge = (E + TILE - 1) / TILE;

    lgcn_init<<<gridNode, BLOCK, 0, stream>>>(ue, ie, out_mean, out_emb0,
                                              bufA, bufB, userElems, totalElems);

    const int STAGES_C = 3;
    float* cur = bufA;
    float* nxt = bufB;
    for (int s = 0; s < STAGES_C; ++s) {
        lgcn_spmm<<<gridEdge, BLOCK, 0, stream>>>(cur, nxt, vals, rows, cols, E);
        // zero old `cur` -> it is next stage's scatter target
        lgcn_accum<<<gridNode, BLOCK, 0, stream>>>(out_mean, nxt, cur, total4,
                                                   s == STAGES_C - 1);
        float* tmp = cur; cur = nxt; nxt = tmp;
    }
}